// GraphConvolution_26706106647237
// MI455X (gfx1250) — compile-verified
//
#include <hip/hip_runtime.h>

// GCN layer for MI455X (gfx1250, wave32):
//   support = x @ W            -> WMMA f32 16x16x4 (exact fp32)
//   out     = relu(segsum(adj_val * support[adj_col]) + bias)
//
// N=100000 nodes, IN_F=256, OUT_F=128, E=3.2M edges.

#define IN_F  256
#define OUT_F 128
#define TILE_ROWS 128     // rows per workgroup (4 waves x 32 rows)
#define KCHUNK 64         // K staged per LDS round
#define SW_STRIDE 288     // pair-interleaved weight row: 128 cols x 2 + 32 pad
                          // 288 % 64 == 32 -> wave b64 reads hit all 64 banks once
#define SX_STRIDE 66      // padded x row stride

typedef __attribute__((ext_vector_type(2))) float v2f;
typedef __attribute__((ext_vector_type(8))) float v8f;

// ---------------------------------------------------------------------------
// Dense GEMM: support[n_rows x 128] = x[n_rows x 256] * w[256 x 128]
// Block: 128 threads (4 waves). Wave w computes rows [base+32w, base+32w+32)
// (two 16-row M tiles) across all 128 output columns (8 N tiles of 16).
// Weight is staged pair-interleaved so each B fragment is ONE ds_load_b64.
// ---------------------------------------------------------------------------
__global__ __launch_bounds__(128)
void gcn_gemm_wmma(const float* __restrict__ x, const float* __restrict__ w,
                   float* __restrict__ support, int n_rows) {
    __shared__ float sW[(KCHUNK / 2) * SW_STRIDE];  // 36 KB
    __shared__ float sX[TILE_ROWS * SX_STRIDE];     // 33.8 KB

    const int tid  = threadIdx.x;
    const int wave = tid >> 5;
    const int lane = tid & 31;
    const int rowbase = blockIdx.x * TILE_ROWS;

    v8f acc0[8] = {};   // rows [wave*32 .. +16)
    v8f acc1[8] = {};   // rows [wave*32+16 .. +32)

    // Fragment addressing per ISA 32-bit layouts (16x4 A, 4x16 B):
    //   lanes 0-15 hold K pair {0,1}; lanes 16-31 hold K pair {2,3}.
    const int m     = lane & 15;
    const int khalf = lane >> 4;   // 0 or 1

    for (int kc = 0; kc < IN_F; kc += KCHUNK) {
        __syncthreads();
        // Stage weight chunk pair-interleaved: sW[(k/2)*S + 2n + (k&1)] = W[kc+k][n].
        // Global read coalesced; LDS write bank-conflict-free.
        for (int i = tid; i < KCHUNK * OUT_F; i += 128) {
            int kr = i >> 7;             // i / 128
            int nc = i & (OUT_F - 1);
            sW[(kr >> 1) * SW_STRIDE + nc * 2 + (kr & 1)] = w[(kc + kr) * OUT_F + nc];
        }
        // Stage x chunk: TILE_ROWS x KCHUNK, coalesced; OOB rows read as 0.
        for (int i = tid; i < TILE_ROWS * KCHUNK; i += 128) {
            int r = i >> 6;              // i / 64
            int c = i & (KCHUNK - 1);
            int row = rowbase + r;
            sX[r * SX_STRIDE + c] =
                (row < n_rows) ? x[(long long)row * IN_F + kc + c] : 0.0f;
        }
        __syncthreads();

        const float* xrow0 = &sX[(wave * 32 + m) * SX_STRIDE + 2 * khalf];
        const float* xrow1 = xrow0 + 16 * SX_STRIDE;

        #pragma unroll
        for (int k0 = 0; k0 < KCHUNK; k0 += 4) {
            v2f a0 = *(const v2f*)(xrow0 + k0);   // one ds_load_b64
            v2f a1 = *(const v2f*)(xrow1 + k0);
            const float* wrow = &sW[((k0 >> 1) + khalf) * SW_STRIDE + m * 2];
            #pragma unroll
            for (int nt = 0; nt < 8; ++nt) {
                v2f b = *(const v2f*)(wrow + nt * 32);  // one ds_load_b64, no movs
                acc0[nt] = __builtin_amdgcn_wmma_f32_16x16x4_f32(
                    false, a0, false, b, (short)0, acc0[nt], false, false);
                acc1[nt] = __builtin_amdgcn_wmma_f32_16x16x4_f32(
                    false, a1, false, b, (short)0, acc1[nt], false, false);
            }
        }
    }

    // C/D layout: VGPR r holds (M=r, N=lane) lanes 0-15; (M=8+r, N=lane-16)
    // lanes 16-31.
    const int mhi = khalf * 8;
    #pragma unroll
    for (int nt = 0; nt < 8; ++nt) {
        #pragma unroll
        for (int r = 0; r < 8; ++r) {
            int row0 = rowbase + wave * 32 + r + mhi;
            int row1 = row0 + 16;
            int col  = nt * 16 + m;
            if (row0 < n_rows)
                support[(long long)row0 * OUT_F + col] = acc0[nt][r];
            if (row1 < n_rows)
                support[(long long)row1 * OUT_F + col] = acc1[nt][r];
        }
    }
}

// ---------------------------------------------------------------------------
// Zero-init the accumulator (harness poisons d_out with 0xAA).
// ---------------------------------------------------------------------------
__global__ __launch_bounds__(256)
void gcn_zero(float* __restrict__ out, int n) {
    int i = blockIdx.x * blockDim.x + threadIdx.x;
    if (i < n) out[i] = 0.0f;
}

// ---------------------------------------------------------------------------
// Edge scatter: one wave32 per edge. Lane l handles features [4l, 4l+4).
// support row gather is a coalesced 512B wave read (support = 51 MB, resident
// in the 192 MB L2); accumulation via global_atomic_add_f32 (L2 atomics).
// ---------------------------------------------------------------------------
__global__ __launch_bounds__(256)
void gcn_scatter(const int* __restrict__ adj_row, const int* __restrict__ adj_col,
                 const float* __restrict__ adj_val,
                 const float* __restrict__ support,
                 float* __restrict__ out, int n_edges) {
    long long gtid = (long long)blockIdx.x * blockDim.x + threadIdx.x;
    int e    = (int)(gtid >> 5);
    int lane = (int)(gtid & 31);
    if (e >= n_edges) return;

    int   row = adj_row[e];
    int   col = adj_col[e];
    float val = adj_val[e];

    const float4* src = (const float4*)(support + (long long)col * OUT_F);
    float4 s = src[lane];

    float* dst = out + (long long)row * OUT_F + lane * 4;
    atomicAdd(dst + 0, val * s.x);
    atomicAdd(dst + 1, val * s.y);
    atomicAdd(dst + 2, val * s.z);
    atomicAdd(dst + 3, val * s.w);
}

// ---------------------------------------------------------------------------
// out = relu(out + bias), in place. OUT_F = 128 -> bias index = i & 127.
// ---------------------------------------------------------------------------
__global__ __launch_bounds__(256)
void gcn_bias_relu(float* __restrict__ out, const float* __restrict__ bias, int n) {
    int i = blockIdx.x * blockDim.x + threadIdx.x;
    if (i < n) {
        float v = out[i] + bias[i & (OUT_F - 1)];
        out[i] = v > 0.0f ? v : 0.0f;
    }
}

// ---------------------------------------------------------------------------
extern "C" void kernel_launch(void* const* d_in, const int* in_sizes, int n_in,
                              void* d_out, int out_size, void* d_ws, size_t ws_size,
                              hipStream_t stream) {
    const float* x       = (const float*)d_in[0];
    const int*   adj_row = (const int*)d_in[1];
    const int*   adj_col = (const int*)d_in[2];
    const float* adj_val = (const float*)d_in[3];
    const float* weight  = (const float*)d_in[4];
    const float* bias    = (const float*)d_in[5];
    float*       out     = (float*)d_out;
    float*       support = (float*)d_ws;   // n_nodes * OUT_F floats (51.2 MB)

    const int n_nodes = in_sizes[0] / IN_F;
    const int n_edges = in_sizes[1];
    const int total   = n_nodes * OUT_F;

    // 1) support = x @ W   (WMMA fp32)
    gcn_gemm_wmma<<<(n_nodes + TILE_ROWS - 1) / TILE_ROWS, 128, 0, stream>>>(
        x, weight, support, n_nodes);

    // 2) out = 0
    gcn_zero<<<(total + 255) / 256, 256, 0, stream>>>(out, total);

    // 3) out += adj_val * support[adj_col]  (segment scatter-add)
    long long sthreads = (long long)n_edges * 32;
    gcn_scatter<<<(unsigned)((sthreads + 255) / 256), 256, 0, stream>>>(
        adj_row, adj_col, adj_val, support, out, n_edges);

    // 4) out = relu(out + bias)
    gcn_bias_relu<<<(total + 255) / 256, 256, 0, stream>>>(out, bias, total);
}